// ItemVectorTransform_9053791060248
// MI455X (gfx1250) — compile-verified
//
#include <hip/hip_runtime.h>

typedef __attribute__((ext_vector_type(2))) float v2f;
typedef __attribute__((ext_vector_type(8))) float v8f;

#define DDIM 50
#define KSLOTS 2048
#define ROWS_PER_BLOCK 128
#define WAVES_PER_BLOCK 8
#define SROW 20  // padded LDS row stride (floats): rows r and r+8 land in disjoint bank groups

__global__ __launch_bounds__(256) void attn_read_f32wmma_kernel(
    const float* __restrict__ x, const float* __restrict__ mem,
    float* __restrict__ out) {
  __shared__ float sS[WAVES_PER_BLOCK][16 * SROW];

  const int tid = threadIdx.x;
  const int lane = tid & 31;
  const int wid = tid >> 5;
  const int lmod = lane & 15;   // column / row-in-tile index
  const int lhalf = lane >> 4;  // which half-wave (selects K offset +2)
  const int blockRow0 = blockIdx.x * ROWS_PER_BLOCK;
  const int row0 = blockRow0 + wid * 16;

  // ---- concat part 1: out[:, 0:50] = x  (uniform trip count: 6400/256 = 25)
  for (int idx = tid; idx < ROWS_PER_BLOCK * DDIM; idx += 256) {
    const int r = idx / DDIM;
    const int c = idx - r * DDIM;
    out[(size_t)(blockRow0 + r) * (2 * DDIM) + c] =
        x[(size_t)(blockRow0 + r) * DDIM + c];
  }

  // ---- preload A1 fragments: x tile, 16 rows x 52 (padded) cols, 13 k4-steps
  // A layout (f32 16x4): lanes 0-15 & 16-31 both hold M=0..15; VGPR0 = K+0 / K+2,
  // VGPR1 = K+1 / K+3 (lane group selects +2).
  v2f afrag[13];
  {
    const float* xrow = x + (size_t)(row0 + lmod) * DDIM;
#pragma unroll
    for (int j = 0; j < 13; ++j) {
      const int d = 4 * j + 2 * lhalf;
      if (j < 12) {
        afrag[j][0] = xrow[d];
        afrag[j][1] = xrow[d + 1];
      } else {
        // d = 48 (lanes<16, valid) or 50 (lanes>=16, pad with zeros)
        const bool ok = (d < DDIM);
        const int dd = ok ? d : (DDIM - 2);  // clamp address, stay in-bounds
        const float t0 = xrow[dd];
        const float t1 = xrow[dd + 1];
        afrag[j][0] = ok ? t0 : 0.0f;
        afrag[j][1] = ok ? t1 : 0.0f;
      }
    }
  }

  // u_read accumulators: 4 N-tiles of 16 d-columns each (50 -> 64 padded)
  v8f acc[4] = {};

  float* sw = sS[wid];

  for (int kt = 0; kt < KSLOTS / 16; ++kt) {
    const int slot0 = kt * 16;

    // ---- GEMM1: S[16 batch x 16 slot] = X_tile . M_tile^T, K-dim = D
    // B layout (f32 4x16): VGPR v, lane-group g -> K row v+2g, N = lane&15 (slot)
    v8f s = {};
    {
      const float* mr = mem + (size_t)(slot0 + lmod) * DDIM;
#pragma unroll
      for (int j = 0; j < 13; ++j) {
        v2f b;
        const int d = 4 * j + 2 * lhalf;
        if (j < 12) {
          b[0] = mr[d];
          b[1] = mr[d + 1];
        } else {
          const bool ok = (d < DDIM);
          const int dd = ok ? d : (DDIM - 2);
          const float t0 = mr[dd];
          const float t1 = mr[dd + 1];
          b[0] = ok ? t0 : 0.0f;
          b[1] = ok ? t1 : 0.0f;
        }
        s = __builtin_amdgcn_wmma_f32_16x16x4_f32(false, afrag[j], false, b,
                                                  (short)0, s, false, false);
      }
    }

    // ---- elementwise exp (layout-independent)
#pragma unroll
    for (int r = 0; r < 8; ++r) s[r] = __expf(s[r]);

    // ---- C-layout -> A-layout transpose of S via per-wave LDS scratch.
    // C layout: VGPR r, lanes<16 -> (M=r, N=lane); lanes>=16 -> (M=r+8, N=lane-16)
#pragma unroll
    for (int r = 0; r < 8; ++r)
      sw[(r + 8 * lhalf) * SROW + lmod] = s[r];

    // A2 frags: lane holds batch row (lmod), K (slot) = 4*j2 + 2*lhalf + {0,1}
    v2f a2[4];
#pragma unroll
    for (int j2 = 0; j2 < 4; ++j2) {
      const int kk = 4 * j2 + 2 * lhalf;
      a2[j2][0] = sw[lmod * SROW + kk];
      a2[j2][1] = sw[lmod * SROW + kk + 1];
    }

    // ---- GEMM2: acc[t] += S . M_tile  (K-dim = 16 slots, N = d-columns)
#pragma unroll
    for (int t = 0; t < 4; ++t) {
      const int dcol = 16 * t + lmod;
      const bool ok = (dcol < DDIM);   // pad d-columns 50..63 with zeros
      const int dc = ok ? dcol : 0;
#pragma unroll
      for (int j2 = 0; j2 < 4; ++j2) {
        const int srow = slot0 + 4 * j2 + 2 * lhalf;
        const float t0 = mem[(size_t)srow * DDIM + dc];
        const float t1 = mem[(size_t)(srow + 1) * DDIM + dc];
        v2f b2;
        b2[0] = ok ? t0 : 0.0f;
        b2[1] = ok ? t1 : 0.0f;
        acc[t] = __builtin_amdgcn_wmma_f32_16x16x4_f32(
            false, a2[j2], false, b2, (short)0, acc[t], false, false);
      }
    }
  }

  // ---- concat part 2: out[:, 50:100] = u_read
  // C layout: VGPR r, lane -> row m = r + 8*lhalf, col dcol = 16*t + lmod
#pragma unroll
  for (int t = 0; t < 4; ++t) {
    const int dcol = 16 * t + lmod;
    if (dcol < DDIM) {
#pragma unroll
      for (int r = 0; r < 8; ++r) {
        const int m = r + 8 * lhalf;
        out[(size_t)(row0 + m) * (2 * DDIM) + DDIM + dcol] = acc[t][r];
      }
    }
  }
}

extern "C" void kernel_launch(void* const* d_in, const int* in_sizes, int n_in,
                              void* d_out, int out_size, void* d_ws,
                              size_t ws_size, hipStream_t stream) {
  const float* x = (const float*)d_in[0];
  const float* mem = (const float*)d_in[1];
  float* out = (float*)d_out;

  const int Brows = in_sizes[0] / DDIM;           // 65536
  const int grid = Brows / ROWS_PER_BLOCK;        // 512 blocks x 256 threads
  attn_read_f32wmma_kernel<<<grid, 256, 0, stream>>>(x, mem, out);
}